// RoiRotate_90640989815509
// MI455X (gfx1250) — compile-verified
//
#include <hip/hip_runtime.h>
#include <stdint.h>

// Problem sizes (fixed by the reference)
#define Bv    16
#define Nv    64
#define Cv    64
#define Hv    160
#define Wv    160
#define WOUT  128
#define OUTH  8
#define CHB   16                 // channels staged/computed per block
#define CAP   2176               // per-channel LDS patch floats (worst case ~2050)
#define ROIS_ELEMS (Bv*Nv*OUTH*WOUT*Cv)   // 67,108,864

typedef unsigned int v4u __attribute__((ext_vector_type(4)));
typedef int          v4i __attribute__((ext_vector_type(4)));
typedef int          v8i __attribute__((ext_vector_type(8)));

#if __has_builtin(__builtin_amdgcn_tensor_load_to_lds)
#define HAVE_TDM 1
#else
#define HAVE_TDM 0
#endif

#if HAVE_TDM
// Issue one 2D TDM tile load: pw x ph f32 tile (row stride 160) -> dense LDS.
// D# packing per CDNA5 ISA ch.8 (groups 0/1; groups 2/3 zero for 2D tiles).
__device__ __forceinline__ void tdm_load_tile_2d(uint64_t gaddr, uint32_t lds_off,
                                                 int pw, int ph) {
    v4u g0;
    g0.x = 1u;                                   // count=1 (valid user D#)
    g0.y = lds_off;                              // LDS byte address
    g0.z = (unsigned int)gaddr;                  // global_addr[31:0]
    g0.w = ((unsigned int)(gaddr >> 32) & 0x01FFFFFFu) | 0x80000000u; // [56:32]|type=2
    v8i g1;
    g1[0] = 0x00020000;                          // data_size=2 (4 bytes)
    g1[1] = (int)(160u << 16);                   // tensor_dim0 = 160 (low16)
    g1[2] = (int)(160u << 16);                   // dim0 hi=0 | tensor_dim1 = 160 (low16)
    g1[3] = (int)((unsigned int)pw << 16);       // dim1 hi=0 | tile_dim0 = pw
    g1[4] = ph & 0xFFFF;                         // tile_dim1 = ph, tile_dim2 = 0
    g1[5] = 160;                                 // tensor_dim0_stride = 160
    g1[6] = 0;
    g1[7] = 0;
    v4i z4 = {0, 0, 0, 0};
#if defined(__clang_major__) && (__clang_major__ >= 23)
    v8i z8 = {0, 0, 0, 0, 0, 0, 0, 0};
    __builtin_amdgcn_tensor_load_to_lds(g0, g1, z4, z4, z8, 0);
#else
    __builtin_amdgcn_tensor_load_to_lds(g0, g1, z4, z4, 0);
#endif
}
#endif

// One block = one (b,n) box x 16-channel slab.
// Phase 1: TDM-stage the (in-bounds) bounding-box patch into LDS.
// Phase 2: bilinear-sample 8x128 outputs from LDS, NT-stream to the output.
extern "C" __global__ __launch_bounds__(256)
void RoiRotate_kernel(const float* __restrict__ feat,
                      const float* __restrict__ boxes,
                      float* __restrict__ out)
{
    extern __shared__ float smem[];          // CHB*CAP floats, LDS offset 0
    const int tid = threadIdx.x;
    const int bn  = blockIdx.x >> 2;         // b*N + n
    const int q   = blockIdx.x & 3;          // channel quarter
    const int c0  = q << 4;
    const int b   = bn >> 6;

    // ---- box geometry (uniform across the block; compiler scalarizes) ----
    const float* bx = boxes + bn * 8;
    float p0x = bx[0]*0.25f, p0y = bx[1]*0.25f;
    float p1x = bx[2]*0.25f, p1y = bx[3]*0.25f;
    float p3x = bx[6]*0.25f, p3y = bx[7]*0.25f;
    float e1x = p1x-p0x, e1y = p1y-p0y;      // top edge -> output x
    float e3x = p3x-p0x, e3y = p3y-p0y;      // left edge -> output y
    float wlen = sqrtf(e1x*e1x + e1y*e1y);
    float hlen = sqrtf(e3x*e3x + e3y*e3y);
    float bwL = fmaxf(wlen, hlen), bhS = fminf(wlen, hlen);
    float box_w = fminf(ceilf(8.0f / bhS * bwL), 128.0f);
    int   bwp   = (int)box_w;

    // ---- in-bounds staged tile = ideal patch ∩ [0,W)x[0,H) ----
    float xmin = p0x + fminf(e1x,0.f) + fminf(e3x,0.f);
    float xmax = p0x + fmaxf(e1x,0.f) + fmaxf(e3x,0.f);
    float ymin = p0y + fminf(e1y,0.f) + fminf(e3y,0.f);
    float ymax = p0y + fmaxf(e1y,0.f) + fmaxf(e3y,0.f);
    int gx0i = (int)floorf(xmin), gy0i = (int)floorf(ymin);
    int gx0c = gx0i < 0 ? 0 : (gx0i > Wv-2 ? Wv-2 : gx0i);
    int gy0c = gy0i < 0 ? 0 : (gy0i > Hv-2 ? Hv-2 : gy0i);
    int xhi = (int)floorf(xmax) + 1; if (xhi > Wv-1) xhi = Wv-1;
    int yhi = (int)floorf(ymax) + 1; if (yhi > Hv-1) yhi = Hv-1;
    int pw = xhi - gx0c + 1;
    if (pw < 2) pw = 2;
    if (pw > Wv - gx0c) pw = Wv - gx0c;
    if (pw > 64) pw = 64;
    int ph = yhi - gy0c + 1;
    if (ph < 2) ph = 2;
    if (ph > Hv - gy0c) ph = Hv - gy0c;
    int phmax = CAP / pw;
    if (ph > phmax) ph = phmax;

    // ---- Phase 1: stage 16 channel patches into LDS ----
    const uint64_t featbase = (uint64_t)(uintptr_t)feat;
    const uint32_t tileoff  = (uint32_t)(gy0c * Wv + gx0c);
#if HAVE_TDM
    if (tid < 32) {                          // wave 0 only (TDM ignores EXEC)
        for (int ch = 0; ch < CHB; ++ch) {
            uint64_t ga = featbase +
                4ull * ((uint64_t)((b * Cv) + c0 + ch) * (Hv * Wv) + tileoff);
            tdm_load_tile_2d(ga, (uint32_t)(ch * CAP) * 4u, pw, ph);
        }
#if __has_builtin(__builtin_amdgcn_s_wait_tensorcnt)
        __builtin_amdgcn_s_wait_tensorcnt(0);
#else
        asm volatile("s_wait_tensorcnt 0x0" ::: "memory");
#endif
    }
    asm volatile("" ::: "memory");
#else
    // Fallback: per-lane async global->LDS (tile is fully in-bounds: no clamps)
    const int area = pw * ph;
    for (int ch = 0; ch < CHB; ++ch) {
        const uint64_t chbase = featbase +
            4ull * ((uint64_t)((b * Cv) + c0 + ch) * (Hv * Wv) + tileoff);
        const uint32_t lbase = (uint32_t)(ch * CAP) * 4u;
        for (int e = tid; e < area; e += 256) {
            int py = e / pw;
            int px = e - py * pw;
            uint64_t ga = chbase + 4ull * (uint32_t)(py * Wv + px);
            uint32_t la = lbase + 4u * (uint32_t)e;
            asm volatile("global_load_async_to_lds_b32 %0, %1, off"
                         :: "v"(la), "v"(ga) : "memory");
        }
    }
    asm volatile("s_wait_asynccnt 0" ::: "memory");
#endif
    __syncthreads();

    // ---- Phase 2: bilinear sample from LDS, NT-stream out ----
    const int   chl  = tid & 15;             // channel within slab
    const int   hid  = tid >> 4;             // position phase (2 x-positions/wave)
    const float invbw = 1.0f / box_w;
    const float* my  = smem + chl * CAP;
    float* outp = out + (size_t)bn * (OUTH * WOUT * Cv) + c0 + chl;

    for (int pos = hid; pos < OUTH * WOUT; pos += 16) {
        int y = pos >> 7;
        int x = pos & 127;
        float u  = (float)x * invbw;
        float v  = (float)y * 0.125f;
        float xs = p0x + v * e3x + u * e1x;
        float ys = p0y + v * e3y + u * e1y;
        float xf = floorf(xs), yf = floorf(ys);
        float fx = xs - xf,   fy = ys - yf;
        int x0 = (int)xf, y0 = (int)yf;
        // per-tap validity (zero border), folded into weights
        float vx0 = (x0 >=  0 && x0 < Wv    ) ? 1.f : 0.f;
        float vx1 = (x0 >= -1 && x0 < Wv - 1) ? 1.f : 0.f;
        float vy0 = (y0 >=  0 && y0 < Hv    ) ? 1.f : 0.f;
        float vy1 = (y0 >= -1 && y0 < Hv - 1) ? 1.f : 0.f;
        int ix_raw = x0 - gx0c;
        int iy_raw = y0 - gy0c;
        int ix = ix_raw < 0 ? 0 : (ix_raw > pw - 2 ? pw - 2 : ix_raw);
        int iy = iy_raw < 0 ? 0 : (iy_raw > ph - 2 ? ph - 2 : iy_raw);
        const float* p = my + iy * pw + ix;
        float v00 = p[0], v01 = p[1], v10 = p[pw], v11 = p[pw + 1];
        // slot selectors: if the raw index over/underflowed the clamp, the
        // valid tap's value sits in the other slot of the loaded 2x2 quad
        // (the displaced tap is out-of-map, weight 0).
        bool sxa = ix_raw > ix;      // x0 column slot (0 normal, 1 if clamped down)
        bool sxb = ix_raw >= 0;      // x1 column slot (1 normal, 0 if clamped up)
        bool sya = iy_raw > iy;      // y0 row slot
        bool syb = iy_raw >= 0;      // y1 row slot
        float a = sya ? (sxa ? v11 : v10) : (sxa ? v01 : v00);   // (y0,x0)
        float bq= sya ? (sxb ? v11 : v10) : (sxb ? v01 : v00);   // (y0,x1)
        float c = syb ? (sxa ? v11 : v10) : (sxa ? v01 : v00);   // (y1,x0)
        float d = syb ? (sxb ? v11 : v10) : (sxb ? v01 : v00);   // (y1,x1)
        float res = a * ((1.f - fx) * (1.f - fy) * vx0 * vy0)
                  + bq* ((      fx) * (1.f - fy) * vx1 * vy0)
                  + c * ((1.f - fx) * (      fy) * vx0 * vy1)
                  + d * ((      fx) * (      fy) * vx1 * vy1);
        res = (x < bwp) ? res : 0.0f;        // pad region -> exact zeros
        __builtin_nontemporal_store(res, outp + (size_t)pos * Cv);
    }

    // widths[B,N] (int32 bits appended after rois)
    if (q == 0 && tid == 0) {
        ((int*)out)[ROIS_ELEMS + bn] = bwp;
    }
}

extern "C" void kernel_launch(void* const* d_in, const int* in_sizes, int n_in,
                              void* d_out, int out_size, void* d_ws, size_t ws_size,
                              hipStream_t stream) {
    const float* feat  = (const float*)d_in[0];   // [B,C,H,W] f32
    const float* boxes = (const float*)d_in[1];   // [B,N,8]   f32
    float* out = (float*)d_out;                   // rois flat + widths bits
    (void)in_sizes; (void)n_in; (void)out_size; (void)d_ws; (void)ws_size;

    const dim3 grid(Bv * Nv * (Cv / CHB));        // 4096 blocks
    const dim3 block(256);                        // 8 waves (wave32)
    const size_t shmem = (size_t)CHB * CAP * sizeof(float);   // 139,264 B LDS
    RoiRotate_kernel<<<grid, block, shmem, stream>>>(feat, boxes, out);
}